// FastLIFNeuron_33784212750507
// MI455X (gfx1250) — compile-verified
//
#include <hip/hip_runtime.h>

// LIF neuron scan: x[B,T,F] fp32 -> spikes[B,T,F] fp32
//   m_t = 0.9*m_{t-1} + x_t ; spike = (m_t > 0.5) ; m_t = spike ? 0 : m_t
// Memory-bound streaming kernel:
//   - coalesced B128 NT loads/stores (one float4 per thread, wave covers 512B)
//   - explicit 8-deep register-rotating load pipeline (8 loads in flight/wave)
//   - global_prefetch_b8 32 timesteps ahead to prime L2 from HBM

typedef __attribute__((ext_vector_type(4))) float fvec4;

#define LIF_T 256
#define LIF_F 1024
#define LIF_F4 (LIF_F / 4)   // 256 float4 columns per (b,t) row

#define PIPE_D 8             // register pipeline depth (loads in flight per wave)
#define PF_T   32            // L2 prefetch distance in timesteps

__device__ __forceinline__ void lif_step(fvec4 xt,
                                         float& m0, float& m1, float& m2, float& m3,
                                         fvec4& s)
{
    const float decay = 0.9f;
    const float thr   = 0.5f;
    m0 = __builtin_fmaf(decay, m0, xt.x);
    m1 = __builtin_fmaf(decay, m1, xt.y);
    m2 = __builtin_fmaf(decay, m2, xt.z);
    m3 = __builtin_fmaf(decay, m3, xt.w);
    s.x = (m0 > thr) ? 1.0f : 0.0f;  m0 = (m0 > thr) ? 0.0f : m0;
    s.y = (m1 > thr) ? 1.0f : 0.0f;  m1 = (m1 > thr) ? 0.0f : m1;
    s.z = (m2 > thr) ? 1.0f : 0.0f;  m2 = (m2 > thr) ? 0.0f : m2;
    s.w = (m3 > thr) ? 1.0f : 0.0f;  m3 = (m3 > thr) ? 0.0f : m3;
}

__global__ __launch_bounds__(128, 4)
void FastLIFNeuron_lif_scan_kernel(const fvec4* __restrict__ x,
                                   fvec4* __restrict__ out)
{
    const int gid = blockIdx.x * blockDim.x + threadIdx.x;   // over B * LIF_F4
    const int b   = gid >> 8;            // / LIF_F4
    const int f4  = gid & (LIF_F4 - 1);  // consecutive tid -> consecutive f4 (coalesced)

    const size_t base = (size_t)b * LIF_T * LIF_F4 + (size_t)f4;
    const fvec4* __restrict__ xp = x   + base;
    fvec4* __restrict__       op = out + base;

    float m0 = 0.0f, m1 = 0.0f, m2 = 0.0f, m3 = 0.0f;

    // ---- prologue: fill the 8-deep register pipeline (8 loads in flight) ----
    fvec4 buf[PIPE_D];
#pragma unroll
    for (int k = 0; k < PIPE_D; ++k) {
        buf[k] = __builtin_nontemporal_load(xp + (size_t)k * LIF_F4);
    }

    // ---- main loop: 31 groups of 8 timesteps (t = 0 .. 247) ----
    // Each step consumes the oldest buffered load and refills its slot with
    // the load for t+8, keeping LOADcnt ~8 deep per wave at all times.
    for (int tg = 0; tg < (LIF_T - PIPE_D) / PIPE_D; ++tg) {
#pragma unroll
        for (int k = 0; k < PIPE_D; ++k) {
            const int t = tg * PIPE_D + k;

            // Prime L2 well ahead of the register pipeline (clamped, branch-free).
            int tp = t + PF_T;
            tp = (tp < LIF_T) ? tp : (LIF_T - 1);
            __builtin_prefetch((const void*)(xp + (size_t)tp * LIF_F4), 0, 0);

            fvec4 xt = buf[k];
            buf[k] = __builtin_nontemporal_load(xp + (size_t)(t + PIPE_D) * LIF_F4);

            fvec4 s;
            lif_step(xt, m0, m1, m2, m3, s);
            __builtin_nontemporal_store(s, op + (size_t)t * LIF_F4);
        }
    }

    // ---- epilogue: drain the last 8 buffered timesteps (t = 248 .. 255) ----
#pragma unroll
    for (int k = 0; k < PIPE_D; ++k) {
        const int t = (LIF_T - PIPE_D) + k;
        fvec4 s;
        lif_step(buf[k], m0, m1, m2, m3, s);
        __builtin_nontemporal_store(s, op + (size_t)t * LIF_F4);
    }
}

extern "C" void kernel_launch(void* const* d_in, const int* in_sizes, int n_in,
                              void* d_out, int out_size, void* d_ws, size_t ws_size,
                              hipStream_t stream)
{
    const fvec4* x = (const fvec4*)d_in[0];
    fvec4* out     = (fvec4*)d_out;

    // x is [B, T=256, F=1024]; derive B from the flat size.
    const int total   = in_sizes[0];
    const int B       = total / (LIF_T * LIF_F);
    const int threads = 128;                       // 4 wave32 per block
    const int nthr    = B * LIF_F4;                // one thread per (b, f4)
    const int blocks  = (nthr + threads - 1) / threads;

    FastLIFNeuron_lif_scan_kernel<<<blocks, threads, 0, stream>>>(x, out);
}